// gaussianDA_33827162423494
// MI455X (gfx1250) — compile-verified
//
#include <hip/hip_runtime.h>

// Problem constants from the reference: B = D = O = 512, all float32.
#define B_DIM 512
#define D_DIM 512
#define O_DIM 512

typedef float v2f __attribute__((ext_vector_type(2)));
typedef float v8f __attribute__((ext_vector_type(8)));

// logits = 2*(x @ (mu*invC)^T) - (x^2 @ invC^T) - c[o],  c[o] = sum_d mu^2*invC
//
// One wave computes a 32x32 output tile using V_WMMA_F32_16X16X4_F32.
// f32 WMMA operand layout (CDNA5 ISA 7.12.2):
//   A 16x4 : lanes 0-15 -> M=lane,   v0=K0,v1=K1 ; lanes 16-31 -> same M, v0=K2,v1=K3
//   B 4x16 : lanes 0-15 -> N=lane,   v0=K0,v1=K1 ; lanes 16-31 -> same N, v0=K2,v1=K3
//   C/D    : lane = N (mod 16), VGPR r -> M = r + 8*(lane/16)
__global__ __launch_bounds__(128) void gaussianDA_wmma_kernel(
    const float* __restrict__ x,       // [B, D] row-major
    const float* __restrict__ mu,      // [O, D] row-major
    const float* __restrict__ invC,    // [O, D] row-major
    float* __restrict__ out)           // [B, O]
{
    const int lane = threadIdx.x & 31;
    const int wave = threadIdx.x >> 5;

    // 4 waves per block -> 64x64 block tile, each wave a 32x32 tile.
    const int M0 = blockIdx.x * 64 + (wave >> 1) * 32;   // batch rows
    const int N0 = blockIdx.y * 64 + (wave & 1) * 32;    // class cols

    const int half = lane >> 4;        // 0: K0/K1 lanes, 1: K2/K3 lanes
    const int idx  = lane & 15;        // M for A loads, N for B loads
    const int kb   = half * 2;         // per-lane K sub-offset within a K=4 step

    v8f acc1[2][2] = {};               // x^2 @ invC^T
    v8f acc2[2][2] = {};               // x   @ (mu*invC)^T
    v2f cpart[2] = {{0.f, 0.f}, {0.f, 0.f}};  // packed partials of c[o] (one v_pk_fma_f32 each)

    const float* xr0 = x    + (size_t)(M0 + idx) * D_DIM + kb;
    const float* xr1 = xr0  + 16 * D_DIM;
    const float* mr0 = mu   + (size_t)(N0 + idx) * D_DIM + kb;
    const float* mr1 = mr0  + 16 * D_DIM;
    const float* ir0 = invC + (size_t)(N0 + idx) * D_DIM + kb;
    const float* ir1 = ir0  + 16 * D_DIM;

    for (int k = 0; k < D_DIM; k += 4) {
        v2f a0 = *(const v2f*)(xr0 + k);   // A tile, M rows M0..M0+15
        v2f a1 = *(const v2f*)(xr1 + k);   // A tile, M rows M0+16..M0+31
        v2f m0 = *(const v2f*)(mr0 + k);   // mu   cols N0..N0+15
        v2f m1 = *(const v2f*)(mr1 + k);   // mu   cols N0+16..N0+31
        v2f i0 = *(const v2f*)(ir0 + k);   // invC cols N0..N0+15
        v2f i1 = *(const v2f*)(ir1 + k);   // invC cols N0+16..N0+31

        v2f a0s = a0 * a0;                 // x^2          (v_pk_mul_f32)
        v2f a1s = a1 * a1;
        v2f w0  = m0 * i0;                 // mu * invC    (v_pk_mul_f32)
        v2f w1  = m1 * i1;

        cpart[0] += m0 * w0;               // mu^2 * invC  (v_pk_fma_f32, stays packed)
        cpart[1] += m1 * w1;

        // term1: (x^2) @ invC^T
        acc1[0][0] = __builtin_amdgcn_wmma_f32_16x16x4_f32(false, a0s, false, i0, (short)0, acc1[0][0], false, false);
        acc1[0][1] = __builtin_amdgcn_wmma_f32_16x16x4_f32(false, a0s, false, i1, (short)0, acc1[0][1], false, false);
        acc1[1][0] = __builtin_amdgcn_wmma_f32_16x16x4_f32(false, a1s, false, i0, (short)0, acc1[1][0], false, false);
        acc1[1][1] = __builtin_amdgcn_wmma_f32_16x16x4_f32(false, a1s, false, i1, (short)0, acc1[1][1], false, false);
        // term2: x @ (mu*invC)^T
        acc2[0][0] = __builtin_amdgcn_wmma_f32_16x16x4_f32(false, a0,  false, w0, (short)0, acc2[0][0], false, false);
        acc2[0][1] = __builtin_amdgcn_wmma_f32_16x16x4_f32(false, a0,  false, w1, (short)0, acc2[0][1], false, false);
        acc2[1][0] = __builtin_amdgcn_wmma_f32_16x16x4_f32(false, a1,  false, w0, (short)0, acc2[1][0], false, false);
        acc2[1][1] = __builtin_amdgcn_wmma_f32_16x16x4_f32(false, a1,  false, w1, (short)0, acc2[1][1], false, false);
    }

    // Complete c[o]: collapse packed pair, then add the other lane-half's K range.
    float c0 = cpart[0].x + cpart[0].y;
    float c1 = cpart[1].x + cpart[1].y;
    float cfull[2];
    cfull[0] = c0 + __shfl_xor(c0, 16, 32);
    cfull[1] = c1 + __shfl_xor(c1, 16, 32);

    // Epilogue: out = 2*acc2 - acc1 - c ; C/D layout -> lane gives N, VGPR r gives M.
    const int n = idx;
#pragma unroll
    for (int mi = 0; mi < 2; ++mi) {
#pragma unroll
        for (int ni = 0; ni < 2; ++ni) {
            const float cf = cfull[ni];
#pragma unroll
            for (int r = 0; r < 8; ++r) {
                const int m   = r + half * 8;
                const int row = M0 + mi * 16 + m;
                const int col = N0 + ni * 16 + n;
                out[(size_t)row * O_DIM + col] =
                    2.0f * acc2[mi][ni][r] - acc1[mi][ni][r] - cf;
            }
        }
    }
}

extern "C" void kernel_launch(void* const* d_in, const int* in_sizes, int n_in,
                              void* d_out, int out_size, void* d_ws, size_t ws_size,
                              hipStream_t stream) {
    (void)in_sizes; (void)n_in; (void)out_size; (void)d_ws; (void)ws_size;
    const float* x    = (const float*)d_in[0];   // [512, 512]
    const float* mu   = (const float*)d_in[1];   // [512, 512]
    const float* invC = (const float*)d_in[2];   // [512, 512]
    float* out        = (float*)d_out;           // [512, 512]

    dim3 grid(B_DIM / 64, O_DIM / 64);           // 8 x 8 blocks
    dim3 block(128);                             // 4 waves -> 64x64 tile/block
    gaussianDA_wmma_kernel<<<grid, block, 0, stream>>>(x, mu, invC, out);
}